// PoincareEmbedding_41386304864903
// MI455X (gfx1250) — compile-verified
//
#include <hip/hip_runtime.h>
#include <hip/hip_bf16.h>

#define SIZE 50000
#define DIM 16
#define BATCH 128
#define GAMMA 1.0f
#define BOUNDARY (1.0f - 1e-5f)

typedef __attribute__((ext_vector_type(2))) float v2f;
typedef __attribute__((ext_vector_type(8))) float v8f;

// ds_swizzle_b32 group-of-32 xor patterns (offset[14:10]=xor, [4:0]=and=0x1f)
#define DS_SWZ_F32(x, imm) \
    __int_as_float(__builtin_amdgcn_ds_swizzle(__float_as_int(x), (imm)))
#define SWZ_XOR16 0x401F
#define SWZ_XOR8  0x201F
#define SWZ_XOR4  0x101F
#define SWZ_XOR2  0x081F
#define SWZ_XOR1  0x041F

// ---------------------------------------------------------------------------
// Kernel 1: gather u rows, compute raw ||u||^2, zero softmax accumulators.
// ---------------------------------------------------------------------------
__global__ __launch_bounds__(128) void poincare_setup(
    const long long* __restrict__ inputs,   // int64 indices
    const float* __restrict__ weight,
    float* __restrict__ u_packed,           // [128][16]
    float* __restrict__ unorm,              // [128] raw sum of squares
    float* __restrict__ row_sums)           // [128] zeroed here
{
    int b = threadIdx.x;
    if (b < BATCH) {
        long long idx = inputs[b];
        const float* src = weight + (size_t)idx * DIM;
        float s = 0.0f;
#pragma unroll
        for (int k = 0; k < DIM; ++k) {
            float w = src[k];
            u_packed[b * DIM + k] = w;
            s = fmaf(w, w, s);
        }
        unorm[b]    = s;
        row_sums[b] = 0.0f;
    }
}

// ---------------------------------------------------------------------------
// Kernel 2: fused WMMA distance + exp pass.
// Block = 256 threads = 8 waves. Wave w computes the 16x16 tile
// rows [16w, 16w+16) x cols [16*blockIdx.x, +16). 3125 blocks cover N=50000.
// ---------------------------------------------------------------------------
__global__ __launch_bounds__(256) void poincare_main(
    const float* __restrict__ weight,
    const float* __restrict__ u_packed,
    const float* __restrict__ unorm,
    float* __restrict__ row_sums,
    float* __restrict__ out)
{
    const int lane = threadIdx.x & 31;
    const int wave = threadIdx.x >> 5;
    const int li   = lane & 15;   // N column within tile / M row for A loads
    const int hi   = lane >> 4;   // half-wave selector
    const int m0   = wave * 16;
    const int col  = blockIdx.x * 16 + li;     // always < 50000 (3125*16)

    // A: u tile (16x4 per step), B: v^T tile (4x16 per step).
    // f32 16x4 A layout: lanes 0-15 -> M=lane, VGPR0/1 = K=k0+2hi, k0+2hi+1.
    v2f a[4], b[4];
    float vpart = 0.0f;
#pragma unroll
    for (int kk = 0; kk < 4; ++kk) {
        const int k = kk * 4 + 2 * hi;
        a[kk] = *(const v2f*)(u_packed + (size_t)(m0 + li) * DIM + k);
        b[kk] = *(const v2f*)(weight   + (size_t)col * DIM + k);
        vpart = fmaf(b[kk].x, b[kk].x, vpart);
        vpart = fmaf(b[kk].y, b[kk].y, vpart);
    }
    // ||v||^2: combine the two K-halves held by opposite half-waves.
    const float vnorm = vpart + DS_SWZ_F32(vpart, SWZ_XOR16);

    // D = A x B + C accumulated over K=16 in four K=4 WMMA steps (pure fp32).
    v8f c = {};
#pragma unroll
    for (int kk = 0; kk < 4; ++kk)
        c = __builtin_amdgcn_wmma_f32_16x16x4_f32(
                /*neg_a=*/false, a[kk], /*neg_b=*/false, b[kk],
                /*c_mod=*/(short)0, c, /*reuse_a=*/false, /*reuse_b=*/false);

    const float dv = 1.0f - fminf(vnorm, BOUNDARY);

#pragma unroll
    for (int j = 0; j < 8; ++j) {
        const int   m  = m0 + j + 8 * hi;       // C/D layout: VGPR j -> M = j + 8*hi
        const float un = unorm[m];
        // sqdist = ||u||^2 + ||v||^2 - 2 u.v  (clamped against rounding)
        const float d  = fmaxf(fmaf(-2.0f, c[j], un + vnorm), 0.0f);
        const float du = 1.0f - fminf(un, BOUNDARY);
        // x = 2*sqdist/((1-||u||^2)(1-||v||^2)) + 1  (fast v_rcp, denom ~= 1)
        const float x  = fmaf(2.0f * d, __builtin_amdgcn_rcpf(du * dv), 1.0f);
        // z = sqrt(x^2-1) via raw v_sqrt_f32
        const float z  = __builtin_amdgcn_sqrtf(fmaxf(fmaf(x, x, -1.0f), 0.0f));
        const float t  = x + z;                  // exp(dist) = x + sqrt(x^2-1)
        float e;
        if (GAMMA == 1.0f) {
            e = __builtin_amdgcn_rcpf(t);        // exp(-1*log t) == 1/t
        } else {
            e = __expf(-GAMMA * __logf(t));
        }

        out[(size_t)m * SIZE + col] = e;

        // Row-sum for softmax denominator: xor-tree over the 16-lane half.
        float s = e;
        s += DS_SWZ_F32(s, SWZ_XOR8);
        s += DS_SWZ_F32(s, SWZ_XOR4);
        s += DS_SWZ_F32(s, SWZ_XOR2);
        s += DS_SWZ_F32(s, SWZ_XOR1);
        if (li == 0)
            atomicAdd(&row_sums[m], s);          // one add per row per block
    }
}

// ---------------------------------------------------------------------------
// Kernel 3a: reciprocal of row sums (128 values; exact division, cost-free).
// ---------------------------------------------------------------------------
__global__ __launch_bounds__(128) void poincare_inv(
    const float* __restrict__ row_sums, float* __restrict__ inv_sums)
{
    int b = threadIdx.x;
    if (b < BATCH) inv_sums[b] = 1.0f / row_sums[b];
}

// ---------------------------------------------------------------------------
// Kernel 3b: normalize, float4 vectorized. 50000 % 4 == 0 so no chunk
// straddles a row. Chunks = 128 * 12500 = 1,600,000 = 6250 blocks * 256.
// ---------------------------------------------------------------------------
__global__ __launch_bounds__(256) void poincare_norm(
    float* __restrict__ out, const float* __restrict__ inv_sums)
{
    const size_t i   = (size_t)blockIdx.x * 256 + threadIdx.x;  // float4 index
    const int    row = (int)(i / (SIZE / 4));
    const float  inv = inv_sums[row];
    float4* p = (float4*)out + i;
    float4  v = *p;
    v.x *= inv; v.y *= inv; v.z *= inv; v.w *= inv;
    *p = v;
}

// ---------------------------------------------------------------------------
extern "C" void kernel_launch(void* const* d_in, const int* in_sizes, int n_in,
                              void* d_out, int out_size, void* d_ws, size_t ws_size,
                              hipStream_t stream)
{
    const long long* inputs = (const long long*)d_in[0];   // int64 (128)
    const float*     weight = (const float*)d_in[1];       // fp32 (50000x16)
    float* out = (float*)d_out;
    float* ws  = (float*)d_ws;

    float* u_packed = ws;            // 2048 floats
    float* unorm    = ws + 2048;     // 128
    float* row_sums = ws + 2176;     // 128
    float* inv_sums = ws + 2304;     // 128

    poincare_setup<<<1, 128, 0, stream>>>(inputs, weight, u_packed, unorm, row_sums);
    poincare_main <<<SIZE / 16, 256, 0, stream>>>(weight, u_packed, unorm, row_sums, out);
    poincare_inv  <<<1, 128, 0, stream>>>(row_sums, inv_sums);
    poincare_norm <<<(BATCH * (SIZE / 4)) / 256, 256, 0, stream>>>(out, inv_sums);
}